// ActiveRotatingFilter_90305982365989
// MI455X (gfx1250) — compile-verified
//
#include <hip/hip_runtime.h>
#include <cstdint>
#include <cstddef>

// Problem constants (from reference setup_inputs)
#define O_DIM   512
#define I_DIM   256
#define NORI    8
#define E_DIM   72          // nOri * kH * kW
#define R_DIM   8
#define IB      128         // rows of I staged per block
#define NTHR    256
#define NBLK    (O_DIM * (I_DIM / IB))           // 1024 blocks
#define CHUNKS  ((IB * E_DIM * 4) / (16 * NTHR)) // 9 x b128 async loads / thread
#define VPT     ((IB * E_DIM) / (4 * NTHR))      // 9 float4 stores / thread / rot

typedef float vfloat4 __attribute__((ext_vector_type(4)));

// out[o, r, i, p] = in[o, i, inv[p, r]],  inv = inverse of the 72-perm idx[:,r]
__global__ __launch_bounds__(NTHR)
void arf_permute_kernel(const float* __restrict__ in,
                        const int*   __restrict__ indices,
                        float*       __restrict__ out)
{
    // 16B alignment required by b128 async-DMA LDS destinations
    __shared__ __attribute__((aligned(16))) float lds_in[IB * E_DIM]; // 36864 B
    __shared__ int sinv[R_DIM * E_DIM];                               //  2304 B

    const int tid = threadIdx.x;
    const int o   = blockIdx.x >> 1;
    const int i0  = (blockIdx.x & 1) * IB;

    // 1) Kick off the async staging of this block's input slab (IB x 72 f32 =
    //    36 KB) into LDS first, so the DMA overlaps the table build below.
    //    CDNA5 path: global_load_async_to_lds_b128 (ASYNCcnt-tracked).
    const float* gsrc = in + ((size_t)o * I_DIM + i0) * E_DIM;   // 288B-aligned
    const unsigned lds_base = (unsigned)(size_t)(&lds_in[0]);    // wave-relative LDS byte addr
#pragma unroll
    for (int k = 0; k < CHUNKS; ++k) {
        int      chunk = tid + k * NTHR;                 // 16-byte chunk id
        unsigned laddr = lds_base + chunk * 16;
        uint64_t gaddr = (uint64_t)(size_t)(gsrc + chunk * 4);
        asm volatile("global_load_async_to_lds_b128 %0, %1, off"
                     :: "v"(laddr), "v"(gaddr)
                     : "memory");
    }

    // 2) Meanwhile build the inverse permutation table in LDS:
    //    sinv[r*72 + p] = e where idx[e][r] == p. indices flat layout is
    //    [e][r] (e = nOri*kH*kW index), values are 1-based. Disjoint LDS range
    //    from the DMA destination, so safe to interleave.
    for (int x = tid; x < E_DIM * R_DIM; x += NTHR) {
        int e = x >> 3;          // x = e*8 + r
        int r = x & 7;
        int p = indices[x] - 1;
        sinv[r * E_DIM + p] = e;
    }

    // 3) Drain this wave's async DMAs, then barrier so every wave sees the
    //    full slab and the complete sinv table.
    asm volatile("s_wait_asynccnt 0x0" ::: "memory");
    __syncthreads();

    // 4) 8 rotations: permuted gather from the LDS row (permutation stays
    //    inside a 72-float row), coalesced 128-bit NT stores. Output per
    //    (o,r) is a contiguous (I,72) slab; row pitch 288 B = 18x16 B so all
    //    float4 stores are aligned. NT keeps the 37.7 MB input resident in L2
    //    while streaming the 302 MB output.
#pragma unroll 1
    for (int r = 0; r < R_DIM; ++r) {
        const int* __restrict__ pinv = &sinv[r * E_DIM];
        float* gout = out + ((size_t)(o * R_DIM + r) * I_DIM + i0) * E_DIM;
#pragma unroll
        for (int k = 0; k < VPT; ++k) {
            int v  = tid + k * NTHR;     // float4 index within slab, < 2304
            int i  = v / 18;             // 18 float4 per 72-float row
            int c  = v - i * 18;
            int p0 = c * 4;
            const float* row = &lds_in[i * E_DIM];
            vfloat4 val;
            val.x = row[pinv[p0 + 0]];
            val.y = row[pinv[p0 + 1]];
            val.z = row[pinv[p0 + 2]];
            val.w = row[pinv[p0 + 3]];
            __builtin_nontemporal_store(val, (vfloat4*)gout + v);
        }
    }
}

extern "C" void kernel_launch(void* const* d_in, const int* in_sizes, int n_in,
                              void* d_out, int out_size, void* d_ws, size_t ws_size,
                              hipStream_t stream)
{
    (void)in_sizes; (void)n_in; (void)out_size; (void)d_ws; (void)ws_size;
    const float* in      = (const float*)d_in[0];   // (512,256,8,3,3) f32
    const int*   indices = (const int*)d_in[1];     // (8,3,3,8) i32, 1-based
    float*       out     = (float*)d_out;           // (4096,2048,3,3) f32

    arf_permute_kernel<<<dim3(NBLK), dim3(NTHR), 0, stream>>>(in, indices, out);
}